// causality_enhanced_message_passing_24489903522211
// MI455X (gfx1250) — compile-verified
//
#include <hip/hip_runtime.h>
#include <hip/hip_bf16.h>
#include <limits.h>

// ---------------------------------------------------------------------------
// Causality-enhanced heterogeneous-graph attention for MI455X (gfx1250).
// f32 end-to-end using V_WMMA_F32_16X16X4_F32 (full precision, no downcast).
// N (=50000) and E (=400000) are multiples of 16 -> all WMMA waves have
// EXEC == all-ones as the ISA requires.
// ---------------------------------------------------------------------------

#define HH   8     // heads
#define DKK  16    // head dim
#define DIM  128   // model dim

typedef __attribute__((ext_vector_type(2))) float v2f;
typedef __attribute__((ext_vector_type(8))) float v8f;

__device__ __forceinline__ v8f wmma_f32_4(v2f a, v2f b, v8f c) {
  // 8 args: (neg_a, A, neg_b, B, c_mod, C, reuse_a, reuse_b)
  return __builtin_amdgcn_wmma_f32_16x16x4_f32(false, a, false, b, (short)0, c,
                                               false, false);
}

// Order-preserving float<->int bijection (involution) for atomicMax on floats.
__device__ __forceinline__ int ord_f32(float f) {
  int i = __float_as_int(f);
  return (i >= 0) ? i : (i ^ 0x7fffffff);
}
__device__ __forceinline__ float unord_f32(int i) {
  return __int_as_float((i >= 0) ? i : (i ^ 0x7fffffff));
}

// ---------------------------------------------------------------------------
// K0: init accumulators
// ---------------------------------------------------------------------------
__global__ __launch_bounds__(256) void init_kernel(float* __restrict__ out,
                                                   float* __restrict__ asum,
                                                   float* __restrict__ casum,
                                                   int* __restrict__ amax,
                                                   int* __restrict__ camax,
                                                   int nOut, int nNH) {
  int i = blockIdx.x * 256 + threadIdx.x;
  if (i < nOut) out[i] = 0.0f;
  if (i < nNH) {
    asum[i] = 0.0f;
    casum[i] = 0.0f;
    amax[i] = INT_MIN;   // decodes below any finite float
    camax[i] = INT_MIN;
  }
}

// ---------------------------------------------------------------------------
// K1: fused 6-way projection GEMM  out[p] = x @ W[p] + b[p]
// grid = (N/16, 6), block = 256 (8 waves; wave w owns cols [16w,16w+16)).
// WMMA f32 16x16x4, K chained over 128 in 32 steps.
// A frag (16x4 f32): lanes 0-15 rows M=lane, K = k0+{0,1}; lanes 16-31 same
// rows, K = k0+{2,3}.  B frag (4x16): same K split, col = lane&15.
// C/D: VGPR r -> row r (lanes<16) / r+8 (lanes>=16), col = lane&15.
// ---------------------------------------------------------------------------
struct ProjArgs {
  const float* x;
  const float* W[6];
  const float* b[6];
  float* out[6];
};

__global__ __launch_bounds__(256) void proj_gemm_kernel(ProjArgs pa) {
  const int lane = threadIdx.x & 31;
  const int wave = threadIdx.x >> 5;   // 0..7 -> output column tile
  const int mt   = blockIdx.x;         // 16-row tile of nodes
  const int p    = blockIdx.y;         // which projection (k,q,v,ck,cq,cv)

  const float* __restrict__ W    = pa.W[p];
  const float* __restrict__ bias = pa.b[p];
  float* __restrict__       outp = pa.out[p];

  const int row  = mt * 16 + (lane & 15);
  const int col  = wave * 16 + (lane & 15);
  const int koff = (lane < 16) ? 0 : 2;

  const float* __restrict__ xrow = pa.x + (size_t)row * DIM;

  v8f c = {0.f, 0.f, 0.f, 0.f, 0.f, 0.f, 0.f, 0.f};
#pragma unroll
  for (int k0 = 0; k0 < DIM; k0 += 4) {
    v2f a, b;
    a.x = xrow[k0 + koff];
    a.y = xrow[k0 + koff + 1];
    b.x = W[(size_t)(k0 + koff) * DIM + col];
    b.y = W[(size_t)(k0 + koff + 1) * DIM + col];
    c = wmma_f32_4(a, b, c);
  }

  const float bv = bias[col];
  const int rbase = mt * 16 + ((lane < 16) ? 0 : 8);
#pragma unroll
  for (int r = 0; r < 8; ++r) {
    outp[(size_t)(rbase + r) * DIM + col] = c[r] + bv;
  }
}

// ---------------------------------------------------------------------------
// K2: per-edge attention logits (VALU; cau_filter is per-edge-type so the B
// matrix varies inside a 16-edge tile -> not WMMA-tileable without sorting).
//   att  = (q[dst]·k[src] + weight * rel_att·k[src]) / 4
//   catt = sum_j ck[src]_j * (F[t,h,j,:]·cq[dst]) / 4
// cau_filter (24KB) + rel_att staged in LDS. Ordered-int atomicMax per dst.
// ---------------------------------------------------------------------------
__global__ __launch_bounds__(256) void edge_att_kernel(
    const float* __restrict__ qb, const float* __restrict__ kb,
    const float* __restrict__ cqb, const float* __restrict__ ckb,
    const float* __restrict__ weight, const float* __restrict__ rel_att,
    const float* __restrict__ cau_filter, const int* __restrict__ src,
    const int* __restrict__ dst, const int* __restrict__ cau,
    float* __restrict__ att, float* __restrict__ catt,
    int* __restrict__ amax, int* __restrict__ camax, int E_) {
  __shared__ float sF[3 * HH * DKK * DKK];  // 6144 f32 = 24KB
  __shared__ float sRA[HH * DKK];
  for (int i = threadIdx.x; i < 3 * HH * DKK * DKK; i += 256)
    sF[i] = cau_filter[i];
  for (int i = threadIdx.x; i < HH * DKK; i += 256) sRA[i] = rel_att[i];
  __syncthreads();

  const int e = blockIdx.x * 256 + threadIdx.x;
  if (e >= E_) return;

  const int s = src[e];
  const int d = dst[e];
  const int t = cau[e];
  const float w = weight[e];
  const float* __restrict__ Ft = sF + t * (HH * DKK * DKK);

#pragma unroll 1
  for (int h = 0; h < HH; ++h) {
    const float* __restrict__ qp  = qb  + (size_t)d * DIM + h * DKK;
    const float* __restrict__ kp  = kb  + (size_t)s * DIM + h * DKK;
    const float* __restrict__ cqp = cqb + (size_t)d * DIM + h * DKK;
    const float* __restrict__ ckp = ckb + (size_t)s * DIM + h * DKK;

    float kq = 0.f, ka = 0.f;
    float cqv[DKK];
#pragma unroll
    for (int j = 0; j < DKK; ++j) {
      const float kv = kp[j];
      kq += qp[j] * kv;
      ka += sRA[h * DKK + j] * kv;
      cqv[j] = cqp[j];
    }
    const float a = (kq + w * ka) * 0.25f;

    float ca = 0.f;
#pragma unroll
    for (int j = 0; j < DKK; ++j) {
      const float* __restrict__ Fr = Ft + h * (DKK * DKK) + j * DKK;
      float g = 0.f;
#pragma unroll
      for (int dd = 0; dd < DKK; ++dd) g += Fr[dd] * cqv[dd];
      ca += ckp[j] * g;
    }
    ca *= 0.25f;

    att[(size_t)e * HH + h]  = a;
    catt[(size_t)e * HH + h] = ca;
    atomicMax(&amax[d * HH + h], ord_f32(a));
    atomicMax(&camax[d * HH + h], ord_f32(ca));
  }
}

// ---------------------------------------------------------------------------
// K3: exp(att - max) in place, accumulate softmax denominators per (dst,h).
// ---------------------------------------------------------------------------
__global__ __launch_bounds__(256) void edge_exp_kernel(
    const int* __restrict__ dst, float* __restrict__ att,
    float* __restrict__ catt, const int* __restrict__ amax,
    const int* __restrict__ camax, float* __restrict__ asum,
    float* __restrict__ casum, int E_) {
  const int e = blockIdx.x * 256 + threadIdx.x;
  if (e >= E_) return;
  const int d = dst[e];
#pragma unroll
  for (int h = 0; h < HH; ++h) {
    const float m  = unord_f32(amax[d * HH + h]);
    const float cm = unord_f32(camax[d * HH + h]);
    const float ex  = __expf(att[(size_t)e * HH + h] - m);
    const float cex = __expf(catt[(size_t)e * HH + h] - cm);
    att[(size_t)e * HH + h]  = ex;
    catt[(size_t)e * HH + h] = cex;
    atomicAdd(&asum[d * HH + h], ex);
    atomicAdd(&casum[d * HH + h], cex);
  }
}

// ---------------------------------------------------------------------------
// K4: WMMA edge messages + softmax-weighted scatter.
// One block (8 waves) per 16-edge tile; wave = head h.
//   val     = v[src]            @ rel_msg[h]      (16x16x16 via 4 WMMAs)
//   cau_val = (cv[src]+t_emb)   @ rel_msg_cau[h]  (16x16x16 via 4 WMMAs)
//   out[dst,h,:] += a*val + ca*cau_val   (atomicAdd)
// ---------------------------------------------------------------------------
__global__ __launch_bounds__(256) void edge_msg_kernel(
    const float* __restrict__ vb, const float* __restrict__ cvb,
    const float* __restrict__ time_emb, const float* __restrict__ rel_msg,
    const float* __restrict__ rel_msg_cau, const int* __restrict__ src,
    const int* __restrict__ dst, const float* __restrict__ att,
    const float* __restrict__ catt, const float* __restrict__ asum,
    const float* __restrict__ casum, float* __restrict__ out) {
  const int lane  = threadIdx.x & 31;
  const int h     = threadIdx.x >> 5;  // wave index = head
  const int ebase = blockIdx.x * 16;

  const int er   = ebase + (lane & 15);      // edge for this lane's A row
  const int sidx = src[er];
  const int koff = (lane < 16) ? 0 : 2;
  const int dcol = lane & 15;

  const float* __restrict__ Bm = rel_msg + h * (DKK * DKK);
  const float* __restrict__ Bc = rel_msg_cau + h * (DKK * DKK);
  const float* __restrict__ vp  = vb  + (size_t)sidx * DIM + h * DKK;
  const float* __restrict__ cvp = cvb + (size_t)sidx * DIM + h * DKK;
  const float* __restrict__ tp  = time_emb + (size_t)er * DKK;

  v8f cval = {0.f, 0.f, 0.f, 0.f, 0.f, 0.f, 0.f, 0.f};
  v8f ccau = {0.f, 0.f, 0.f, 0.f, 0.f, 0.f, 0.f, 0.f};
#pragma unroll
  for (int k0 = 0; k0 < DKK; k0 += 4) {
    v2f a, b;
    a.x = vp[k0 + koff];
    a.y = vp[k0 + koff + 1];
    b.x = Bm[(k0 + koff) * DKK + dcol];
    b.y = Bm[(k0 + koff + 1) * DKK + dcol];
    cval = wmma_f32_4(a, b, cval);

    v2f a2, b2;
    a2.x = cvp[k0 + koff] + tp[k0 + koff];
    a2.y = cvp[k0 + koff + 1] + tp[k0 + koff + 1];
    b2.x = Bc[(k0 + koff) * DKK + dcol];
    b2.y = Bc[(k0 + koff + 1) * DKK + dcol];
    ccau = wmma_f32_4(a2, b2, ccau);
  }

  // C/D layout: VGPR r -> edge ebase + r (lanes<16) / ebase + r + 8 (>=16).
  const int radd = (lane < 16) ? 0 : 8;
#pragma unroll
  for (int r = 0; r < 8; ++r) {
    const int e = ebase + r + radd;
    const int d = dst[e];
    const float a  = att[(size_t)e * HH + h]  / asum[d * HH + h];
    const float ca = catt[(size_t)e * HH + h] / casum[d * HH + h];
    atomicAdd(&out[(size_t)d * DIM + h * DKK + dcol],
              a * cval[r] + ca * ccau[r]);
  }
}

// ---------------------------------------------------------------------------
// K5: ReLU in place
// ---------------------------------------------------------------------------
__global__ __launch_bounds__(256) void relu_kernel(float* __restrict__ out,
                                                   int n) {
  const int i = blockIdx.x * 256 + threadIdx.x;
  if (i < n) out[i] = fmaxf(out[i], 0.0f);
}

// ---------------------------------------------------------------------------
// Host launcher
// ---------------------------------------------------------------------------
extern "C" void kernel_launch(void* const* d_in, const int* in_sizes, int n_in,
                              void* d_out, int out_size, void* d_ws,
                              size_t ws_size, hipStream_t stream) {
  // setup_inputs() flat order:
  // 0:x 1:weight 2:time_emb 3:src 4:dst 5:cau_types
  // 6..17: Wk,bk,Wq,bq,Wv,bv,Wck,bck,Wcq,bcq,Wcv,bcv
  // 18:rel_att 19:rel_msg 20:rel_msg_cau 21:cau_filter
  const float* x        = (const float*)d_in[0];
  const float* weight   = (const float*)d_in[1];
  const float* time_emb = (const float*)d_in[2];
  const int*   src      = (const int*)d_in[3];
  const int*   dst      = (const int*)d_in[4];
  const int*   cau      = (const int*)d_in[5];
  const float* Wk  = (const float*)d_in[6];
  const float* bk  = (const float*)d_in[7];
  const float* Wq  = (const float*)d_in[8];
  const float* bq  = (const float*)d_in[9];
  const float* Wv  = (const float*)d_in[10];
  const float* bv  = (const float*)d_in[11];
  const float* Wck = (const float*)d_in[12];
  const float* bck = (const float*)d_in[13];
  const float* Wcq = (const float*)d_in[14];
  const float* bcq = (const float*)d_in[15];
  const float* Wcv = (const float*)d_in[16];
  const float* bcv = (const float*)d_in[17];
  const float* rel_att     = (const float*)d_in[18];
  const float* rel_msg     = (const float*)d_in[19];
  const float* rel_msg_cau = (const float*)d_in[20];
  const float* cau_filter  = (const float*)d_in[21];

  const int N_ = in_sizes[0] / DIM;  // 50000 (multiple of 16)
  const int E_ = in_sizes[1];        // 400000 (multiple of 16)

  float* out = (float*)d_out;

  // Workspace layout (floats):
  //   6 projections (N*128 each), att (E*8), catt (E*8),
  //   asum/casum (N*8 each), amax/camax (N*8 ints each).  ~186 MB total.
  float* ws = (float*)d_ws;
  const size_t NP = (size_t)N_ * DIM;
  float* kbuf  = ws;
  float* qbuf  = kbuf + NP;
  float* vbuf  = qbuf + NP;
  float* ckbuf = vbuf + NP;
  float* cqbuf = ckbuf + NP;
  float* cvbuf = cqbuf + NP;
  float* attb  = cvbuf + NP;              // E*8
  float* cattb = attb + (size_t)E_ * HH;  // E*8
  float* asum  = cattb + (size_t)E_ * HH; // N*8
  float* casum = asum + (size_t)N_ * HH;  // N*8
  int*   amax  = (int*)(casum + (size_t)N_ * HH);
  int*   camax = amax + (size_t)N_ * HH;

  // K0: init
  {
    const int nOut = N_ * DIM;
    const int nNH = N_ * HH;
    const int nmax = nOut;  // nOut > nNH
    init_kernel<<<(nmax + 255) / 256, 256, 0, stream>>>(out, asum, casum, amax,
                                                        camax, nOut, nNH);
  }

  // K1: six projection GEMMs via WMMA f32 16x16x4
  {
    ProjArgs pa;
    pa.x = x;
    pa.W[0] = Wk;  pa.b[0] = bk;  pa.out[0] = kbuf;
    pa.W[1] = Wq;  pa.b[1] = bq;  pa.out[1] = qbuf;
    pa.W[2] = Wv;  pa.b[2] = bv;  pa.out[2] = vbuf;
    pa.W[3] = Wck; pa.b[3] = bck; pa.out[3] = ckbuf;
    pa.W[4] = Wcq; pa.b[4] = bcq; pa.out[4] = cqbuf;
    pa.W[5] = Wcv; pa.b[5] = bcv; pa.out[5] = cvbuf;
    dim3 grid(N_ / 16, 6);
    proj_gemm_kernel<<<grid, 256, 0, stream>>>(pa);
  }

  // K2: attention logits + per-dst running max
  edge_att_kernel<<<(E_ + 255) / 256, 256, 0, stream>>>(
      qbuf, kbuf, cqbuf, ckbuf, weight, rel_att, cau_filter, src, dst, cau,
      attb, cattb, amax, camax, E_);

  // K3: exp + denominators
  edge_exp_kernel<<<(E_ + 255) / 256, 256, 0, stream>>>(
      dst, attb, cattb, amax, camax, asum, casum, E_);

  // K4: WMMA edge messages + weighted scatter-add
  edge_msg_kernel<<<E_ / 16, 256, 0, stream>>>(
      vbuf, cvbuf, time_emb, rel_msg, rel_msg_cau, src, dst, attb, cattb, asum,
      casum, out);

  // K5: ReLU
  relu_kernel<<<(N_ * DIM + 255) / 256, 256, 0, stream>>>(out, N_ * DIM);
}